// Hypercomplex4DBNActivation_17033840296047
// MI455X (gfx1250) — compile-verified
//
#include <hip/hip_runtime.h>
#include <math.h>

// ---------------------------------------------------------------------------
// Hypercomplex (quaternion) whitening BN + ELU for MI455X (gfx1250, wave32).
//
// x: [32,56,56,256] fp32 = 25.7M elems (~103 MB).  Memory-bound:
//   pass1 read 103MB (stats) + pass3 read 103MB (L2 hit: 103MB < 192MB L2)
//   + 103MB NT stores  =>  ~206MB HBM  =>  ~9us @ 23.3 TB/s.
// Compute (~8 VALU/elem incl. ELU exp) sits under that roofline.
// The only real matmuls (64x per-channel 4x4 M = G * V^{-1/2}) are done with
// V_WMMA_F32_16X16X4_F32, 4 quaternion channels per WMMA (diagonal blocks).
// ---------------------------------------------------------------------------

typedef __attribute__((ext_vector_type(2))) float v2f;
typedef __attribute__((ext_vector_type(8))) float v8f;

#define HBN_P   100352      // 32*56*56 spatial positions
#define HBN_C   256
#define HBN_EPS 1e-6f

// workspace layout (floats)
#define WS_STATS 0          // 14 stats x 64 d   (sum r,i,j,k; 10 moments)
#define WS_M     1024       // 64 x 16  combined 4x4 matrices
#define WS_BIAS  2048       // 64 x 4   combined biases

__global__ void hbn_init(float* __restrict__ ws) {
  int t = blockIdx.x * blockDim.x + threadIdx.x;
  if (t < 14 * 64) ws[WS_STATS + t] = 0.0f;
}

// ---------------- pass 1: sums + second moments -----------------------------
__global__ __launch_bounds__(256) void hbn_stats(const float* __restrict__ x,
                                                 float* __restrict__ ws) {
  const int d  = threadIdx.x & 63;   // quaternion channel
  const int pg = threadIdx.x >> 6;   // 4 position slots per block
  float acc[14];
#pragma unroll
  for (int q = 0; q < 14; ++q) acc[q] = 0.0f;

  const int pstep = gridDim.x * 4;
  for (int p = blockIdx.x * 4 + pg; p < HBN_P; p += pstep) {
    const float* px = x + (size_t)p * HBN_C + d;
    __builtin_prefetch(px + (size_t)pstep * HBN_C, 0, 3);  // global_prefetch_b8
    float r = px[0], i = px[64], j = px[128], k = px[192]; // coalesced 128B/wave
    acc[0] += r;  acc[1] += i;  acc[2] += j;  acc[3] += k;
    acc[4] += r * r;  acc[5] += r * i;  acc[6] += r * j;  acc[7] += r * k;
    acc[8] += i * i;  acc[9] += i * j;  acc[10] += i * k;
    acc[11] += j * j; acc[12] += j * k; acc[13] += k * k;
  }

  __shared__ float red[4][64][14];
#pragma unroll
  for (int q = 0; q < 14; ++q) red[pg][d][q] = acc[q];
  __syncthreads();
  if (pg == 0) {
#pragma unroll
    for (int q = 0; q < 14; ++q) {
      float v = red[0][d][q] + red[1][d][q] + red[2][d][q] + red[3][d][q];
      atomicAdd(&ws[WS_STATS + q * 64 + d], v);
    }
  }
}

// ---------------- pass 2: eigensolve + WMMA fold M = G * W ------------------
__global__ __launch_bounds__(64) void hbn_solve(
    const float* __restrict__ beta,
    const float* __restrict__ g_rr, const float* __restrict__ g_ri,
    const float* __restrict__ g_rj, const float* __restrict__ g_rk,
    const float* __restrict__ g_ii, const float* __restrict__ g_ij,
    const float* __restrict__ g_ik, const float* __restrict__ g_jj,
    const float* __restrict__ g_jk, const float* __restrict__ g_kk,
    float* __restrict__ ws) {
  const int d    = threadIdx.x;      // one quaternion channel per thread
  const int lane = d & 31;
  const int wave = d >> 5;
  const float invP = 1.0f / (float)HBN_P;

  float s[14];
#pragma unroll
  for (int q = 0; q < 14; ++q) s[q] = ws[WS_STATS + q * 64 + d];
  float mean[4] = { s[0] * invP, s[1] * invP, s[2] * invP, s[3] * invP };

  // covariance V = E[xy] - mx*my, + EPS on diagonal
  float V[4][4];
  V[0][0] = s[4] * invP - mean[0] * mean[0] + HBN_EPS;
  V[0][1] = V[1][0] = s[5] * invP - mean[0] * mean[1];
  V[0][2] = V[2][0] = s[6] * invP - mean[0] * mean[2];
  V[0][3] = V[3][0] = s[7] * invP - mean[0] * mean[3];
  V[1][1] = s[8] * invP - mean[1] * mean[1] + HBN_EPS;
  V[1][2] = V[2][1] = s[9] * invP - mean[1] * mean[2];
  V[1][3] = V[3][1] = s[10] * invP - mean[1] * mean[3];
  V[2][2] = s[11] * invP - mean[2] * mean[2] + HBN_EPS;
  V[2][3] = V[3][2] = s[12] * invP - mean[2] * mean[3];
  V[3][3] = s[13] * invP - mean[3] * mean[3] + HBN_EPS;

  // cyclic Jacobi eigensolve of 4x4 SPD (fully unrolled -> registers)
  float U[4][4] = {{1,0,0,0},{0,1,0,0},{0,0,1,0},{0,0,0,1}};
  const int PP[6] = {0,0,0,1,1,2};
  const int QQ[6] = {1,2,3,2,3,3};
#pragma unroll
  for (int sweep = 0; sweep < 6; ++sweep) {
#pragma unroll
    for (int r6 = 0; r6 < 6; ++r6) {
      const int p = PP[r6], q = QQ[r6];
      float apq  = V[p][q];
      float diff = V[q][q] - V[p][p];
      float c, sn;
      if (fabsf(apq) > 1e-20f) {
        float theta = 0.5f * diff / apq;
        float t = 1.0f / (fabsf(theta) + sqrtf(theta * theta + 1.0f));
        t = (theta >= 0.0f) ? t : -t;
        c  = 1.0f / sqrtf(t * t + 1.0f);
        sn = t * c;
      } else { c = 1.0f; sn = 0.0f; }
#pragma unroll
      for (int k2 = 0; k2 < 4; ++k2) {       // rows: J^T * V
        float vpk = V[p][k2], vqk = V[q][k2];
        V[p][k2] = c * vpk - sn * vqk;
        V[q][k2] = sn * vpk + c * vqk;
      }
#pragma unroll
      for (int k2 = 0; k2 < 4; ++k2) {       // cols: (.)*J, and U = U*J
        float vkp = V[k2][p], vkq = V[k2][q];
        V[k2][p] = c * vkp - sn * vkq;
        V[k2][q] = sn * vkp + c * vkq;
        float ukp = U[k2][p], ukq = U[k2][q];
        U[k2][p] = c * ukp - sn * ukq;
        U[k2][q] = sn * ukp + c * ukq;
      }
    }
  }

  // W = U diag(lam^{-1/2}) U^T
  float isq[4];
#pragma unroll
  for (int b = 0; b < 4; ++b) isq[b] = 1.0f / sqrtf(fmaxf(V[b][b], 1e-12f));
  float Wm[4][4];
#pragma unroll
  for (int a = 0; a < 4; ++a)
#pragma unroll
    for (int cc = 0; cc < 4; ++cc)
      Wm[a][cc] = U[a][0]*isq[0]*U[cc][0] + U[a][1]*isq[1]*U[cc][1]
                + U[a][2]*isq[2]*U[cc][2] + U[a][3]*isq[3]*U[cc][3];

  // symmetric affine G
  float G[4][4];
  { float vrr=g_rr[d], vri=g_ri[d], vrj=g_rj[d], vrk=g_rk[d];
    float vii=g_ii[d], vij=g_ij[d], vik=g_ik[d];
    float vjj=g_jj[d], vjk=g_jk[d], vkk=g_kk[d];
    G[0][0]=vrr; G[0][1]=vri; G[0][2]=vrj; G[0][3]=vrk;
    G[1][0]=vri; G[1][1]=vii; G[1][2]=vij; G[1][3]=vik;
    G[2][0]=vrj; G[2][1]=vij; G[2][2]=vjj; G[2][3]=vjk;
    G[3][0]=vrk; G[3][1]=vik; G[3][2]=vjk; G[3][3]=vkk; }

  // stage G, W into LDS row-major, then M = G*W via V_WMMA_F32_16X16X4_F32,
  // 4 quaternion channels per WMMA (useful output = diagonal 4x4 blocks).
  __shared__ float sG[64][16], sW[64][16];
#pragma unroll
  for (int e = 0; e < 4; ++e)
#pragma unroll
    for (int k2 = 0; k2 < 4; ++k2) { sG[d][e*4+k2] = G[e][k2]; sW[d][e*4+k2] = Wm[e][k2]; }
  __syncthreads();

  __shared__ float sD[2][16][16];
  const int m16   = lane & 15;
  const int khalf = (lane < 16) ? 0 : 2;     // ISA A-layout: K 0/1 low lanes, 2/3 high
  const int dd    = m16 >> 2;                // channel within 4-pack
  const int ec    = m16 & 3;                 // row e (for A) / col c (for B)

  for (int sub = 0; sub < 8; ++sub) {
    const int d_base = wave * 32 + sub * 4;
    v2f Aop, Bop;
    Aop.x = sG[d_base + dd][ec * 4 + khalf];        // A[m=4dd+e][k]
    Aop.y = sG[d_base + dd][ec * 4 + khalf + 1];
    Bop.x = sW[d_base + dd][khalf * 4 + ec];        // B[k][n=4dd+c]
    Bop.y = sW[d_base + dd][(khalf + 1) * 4 + ec];
    v8f Dm = {0.f,0.f,0.f,0.f,0.f,0.f,0.f,0.f};
    Dm = __builtin_amdgcn_wmma_f32_16x16x4_f32(false, Aop, false, Bop,
                                               (short)0, Dm, false, false);
#pragma unroll
    for (int g = 0; g < 8; ++g) {            // D layout: VGPR g = rows g / g+8
      int row = (lane < 16) ? g : (g + 8);
      sD[wave][row][m16] = Dm[g];
    }
    __syncthreads();
    if (d >= d_base && d < d_base + 4) {     // owner extracts its diagonal block
      const int dm = d & 3;
      float Mv[16];
#pragma unroll
      for (int e = 0; e < 4; ++e)
#pragma unroll
        for (int cc = 0; cc < 4; ++cc) Mv[e*4+cc] = sD[wave][4*dm+e][4*dm+cc];
#pragma unroll
      for (int q = 0; q < 16; ++q) ws[WS_M + d*16 + q] = Mv[q];
#pragma unroll
      for (int e = 0; e < 4; ++e) {          // bias absorbs mean and beta
        float b = beta[e*64 + d]
                - (Mv[e*4+0]*mean[0] + Mv[e*4+1]*mean[1]
                 + Mv[e*4+2]*mean[2] + Mv[e*4+3]*mean[3]);
        ws[WS_BIAS + d*4 + e] = b;
      }
    }
    __syncthreads();
  }
}

// ---------------- pass 3: y = M*x + b, ELU (streaming, L2-hit reads) --------
__global__ __launch_bounds__(256) void hbn_apply(const float* __restrict__ x,
                                                 const float* __restrict__ ws,
                                                 float* __restrict__ out) {
  const int d  = threadIdx.x & 63;
  const int pg = threadIdx.x >> 6;

  float M[16], bb[4];
#pragma unroll
  for (int q = 0; q < 16; ++q) M[q] = ws[WS_M + d*16 + q];   // L2 broadcast
#pragma unroll
  for (int e = 0; e < 4; ++e) bb[e] = ws[WS_BIAS + d*4 + e];

  const int pstep = gridDim.x * 4;
  for (int p = blockIdx.x * 4 + pg; p < HBN_P; p += pstep) {
    const float* px = x + (size_t)p * HBN_C + d;
    __builtin_prefetch(px + (size_t)pstep * HBN_C, 0, 3);
    float r = px[0], i = px[64], j = px[128], k = px[192];
    float y0 = fmaf(M[0],  r, fmaf(M[1],  i, fmaf(M[2],  j, fmaf(M[3],  k, bb[0]))));
    float y1 = fmaf(M[4],  r, fmaf(M[5],  i, fmaf(M[6],  j, fmaf(M[7],  k, bb[1]))));
    float y2 = fmaf(M[8],  r, fmaf(M[9],  i, fmaf(M[10], j, fmaf(M[11], k, bb[2]))));
    float y3 = fmaf(M[12], r, fmaf(M[13], i, fmaf(M[14], j, fmaf(M[15], k, bb[3]))));
    y0 = (y0 > 0.0f) ? y0 : (__expf(y0) - 1.0f);   // ELU(alpha=1)
    y1 = (y1 > 0.0f) ? y1 : (__expf(y1) - 1.0f);
    y2 = (y2 > 0.0f) ? y2 : (__expf(y2) - 1.0f);
    y3 = (y3 > 0.0f) ? y3 : (__expf(y3) - 1.0f);
    float* po = out + (size_t)p * HBN_C + d;
    __builtin_nontemporal_store(y0, po);           // don't evict x from L2
    __builtin_nontemporal_store(y1, po + 64);
    __builtin_nontemporal_store(y2, po + 128);
    __builtin_nontemporal_store(y3, po + 192);
  }
}

// ---------------------------------------------------------------------------
extern "C" void kernel_launch(void* const* d_in, const int* in_sizes, int n_in,
                              void* d_out, int out_size, void* d_ws, size_t ws_size,
                              hipStream_t stream) {
  const float* x    = (const float*)d_in[0];
  const float* beta = (const float*)d_in[1];
  const float* grr  = (const float*)d_in[2];
  const float* gri  = (const float*)d_in[3];
  const float* grj  = (const float*)d_in[4];
  const float* grk  = (const float*)d_in[5];
  const float* gii  = (const float*)d_in[6];
  const float* gij  = (const float*)d_in[7];
  const float* gik  = (const float*)d_in[8];
  const float* gjj  = (const float*)d_in[9];
  const float* gjk  = (const float*)d_in[10];
  const float* gkk  = (const float*)d_in[11];
  float* out = (float*)d_out;
  float* ws  = (float*)d_ws;

  hbn_init <<<4,    256, 0, stream>>>(ws);
  hbn_stats<<<1024, 256, 0, stream>>>(x, ws);
  hbn_solve<<<1,    64,  0, stream>>>(beta, grr, gri, grj, grk,
                                      gii, gij, gik, gjj, gjk, gkk, ws);
  hbn_apply<<<2048, 256, 0, stream>>>(x, ws, out);
}